// LearnedKernel_79439715107057
// MI455X (gfx1250) — compile-verified
//
#include <hip/hip_runtime.h>

typedef float v2f __attribute__((ext_vector_type(2)));
typedef float v8f __attribute__((ext_vector_type(8)));

#define NPTS 8192
#define NTILE 512  // NPTS / 16

// ---- workspace layout (float offsets into d_ws) ----
#define OFF_DIST   0                      // double[2] occupies W[0..3]: distS, distS2
#define OFF_BAND   4                      // float[3] per-band accumulators
#define OFF_SUMB   8                      // float[4]: sumU, sumU2, sumUX, sumUY (base)
#define OFF_SUMT   12                     // float[4]: same (target)
#define OFF_PARAMS 16                     // float[8]: Ub,Ut,mubx,muby,mutx,muty,wvar
#define OFF_SP_B   64
#define OFF_SP_T   (OFF_SP_B + 2*NPTS)
#define OFF_U_B    (OFF_SP_T + 2*NPTS)
#define OFF_U_T    (OFF_U_B + NPTS)
#define OFF_W_B    (OFF_U_T + NPTS)
#define OFF_W_T    (OFF_W_B + NPTS)
#define OFF_A4C_B  (OFF_W_T + NPTS)
#define OFF_B4C_B  (OFF_A4C_B + 4*NPTS)
#define OFF_A4C_T  (OFF_B4C_B + 4*NPTS)
#define OFF_B4C_T  (OFF_A4C_T + 4*NPTS)
#define OFF_A4U_B  (OFF_B4C_T + 4*NPTS)
#define OFF_B4U_T  (OFF_A4U_B + 4*NPTS)

__device__ __forceinline__ float fexp(float x) {
  return __builtin_amdgcn_exp2f(x * 1.44269504088896340736f);
}
__device__ __forceinline__ float fsoftplus(float x) {
  return (x > 20.f) ? x
       : 0.69314718055994530942f * __builtin_amdgcn_logf(1.f + fexp(x));
}

__global__ void k_init(float* W) {
  int i = threadIdx.x;
  if (i < 24) W[i] = 0.f;
}

// per-point: scale, weight MLP (2->32->1), softplus; wave-reduced atomic sums
__global__ __launch_bounds__(256)
void k_prep(const float* __restrict__ base, const float* __restrict__ targ,
            const float* __restrict__ log_scale,
            const float* __restrict__ w1, const float* __restrict__ b1,
            const float* __restrict__ w2, const float* __restrict__ b2,
            float* W) {
  int gid = blockIdx.x * 256 + threadIdx.x;    // [0, 2*NPTS)
  bool isB = gid < NPTS;                        // blocks are cloud-uniform (NPTS % 256 == 0)
  int idx = isB ? gid : gid - NPTS;
  float sx = fexp(log_scale[0]), sy = fexp(log_scale[1]);
  const float* P = isB ? base : targ;
  float px = P[2*idx] * sx, py = P[2*idx+1] * sy;
  float logit = b2[0];
  #pragma unroll
  for (int k = 0; k < 32; ++k) {
    float h = fmaxf(px * w1[k] + py * w1[32 + k] + b1[k], 0.f);
    logit += h * w2[k];
  }
  float u = fsoftplus(logit) + 1e-6f;
  float* SP = W + (isB ? OFF_SP_B : OFF_SP_T);
  SP[2*idx] = px; SP[2*idx+1] = py;
  (W + (isB ? OFF_U_B : OFF_U_T))[idx] = u;
  float s0 = u, s1 = u * u, s2 = u * px, s3 = u * py;
  for (int o = 16; o > 0; o >>= 1) {
    s0 += __shfl_down(s0, o); s1 += __shfl_down(s1, o);
    s2 += __shfl_down(s2, o); s3 += __shfl_down(s3, o);
  }
  if ((threadIdx.x & 31) == 0) {
    float* S = W + (isB ? OFF_SUMB : OFF_SUMT);
    atomicAdd(&S[0], s0); atomicAdd(&S[1], s1);
    atomicAdd(&S[2], s2); atomicAdd(&S[3], s3);
  }
}

__global__ void k_means(float* W) {
  if (threadIdx.x != 0) return;
  float Ub = W[OFF_SUMB], U2b = W[OFF_SUMB+1];
  float Ut = W[OFF_SUMT], U2t = W[OFF_SUMT+1];
  float n = (float)NPTS;
  float wv = (U2b / (Ub*Ub) - 1.f/n) / (n - 1.f)
           + (U2t / (Ut*Ut) - 1.f/n) / (n - 1.f);
  float* P = W + OFF_PARAMS;
  P[0] = Ub; P[1] = Ut;
  P[2] = W[OFF_SUMB+2] / Ub; P[3] = W[OFF_SUMB+3] / Ub;
  P[4] = W[OFF_SUMT+2] / Ut; P[5] = W[OFF_SUMT+3] / Ut;
  P[6] = wv;
}

// normalized weights + per-point rank-4 embeddings:
// A-row = [x, y, |p|^2, 1] ; B-col = [-2x, -2y, 1, |q|^2]  =>  A.B = d^2
__global__ __launch_bounds__(256)
void k_build(float* W) {
  int gid = blockIdx.x * 256 + threadIdx.x;
  bool isB = gid < NPTS;
  int idx = isB ? gid : gid - NPTS;
  const float* P = W + OFF_PARAMS;
  float U  = isB ? P[0] : P[1];
  float mx = isB ? P[2] : P[4];
  float my = isB ? P[3] : P[5];
  const float* SP = W + (isB ? OFF_SP_B : OFF_SP_T);
  float px = SP[2*idx], py = SP[2*idx+1];
  float u = (W + (isB ? OFF_U_B : OFF_U_T))[idx];
  (W + (isB ? OFF_W_B : OFF_W_T))[idx] = u / U;
  float cx = px - mx, cy = py - my, n2c = cx*cx + cy*cy;
  float* A4 = W + (isB ? OFF_A4C_B : OFF_A4C_T) + 4*idx;
  A4[0] = cx; A4[1] = cy; A4[2] = n2c; A4[3] = 1.f;
  float* B4 = W + (isB ? OFF_B4C_B : OFF_B4C_T) + 4*idx;
  B4[0] = -2.f*cx; B4[1] = -2.f*cy; B4[2] = 1.f; B4[3] = n2c;
  float n2u = px*px + py*py;
  if (isB) {
    float* A = W + OFF_A4U_B + 4*idx;
    A[0] = px; A[1] = py; A[2] = n2u; A[3] = 1.f;
  } else {
    float* B = W + OFF_B4U_T + 4*idx;
    B[0] = -2.f*px; B[1] = -2.f*py; B[2] = 1.f; B[3] = n2u;
  }
}

// main kernel: one V_WMMA_F32_16X16X4_F32 per 16x16 tile of squared distances
__global__ __launch_bounds__(256)
void k_tiles(float* W, const float* __restrict__ log_sigmas) {
  // sigma band coefficients: exp(-d2/(2s^2)) = exp2(-d2 * c)
  float s0v = fexp(log_sigmas[0]), s1v = fexp(log_sigmas[1]), s2v = fexp(log_sigmas[2]);
  float c0 = 1.44269504088896340736f / (2.f * s0v * s0v);
  float c1 = 1.44269504088896340736f / (2.f * s1v * s1v);
  float c2 = 1.44269504088896340736f / (2.f * s2v * s2v);

  const int lane = threadIdx.x & 31;
  const int hf = lane >> 4, ln = lane & 15;
  const int wavesPerBlock = blockDim.x >> 5;
  const int waveId = blockIdx.x * wavesPerBlock + (threadIdx.x >> 5);
  const int numWaves = gridDim.x * wavesPerBlock;
  const int TILES1 = NTILE * NTILE;
  const int TILES = 3 * TILES1;

  float acc0 = 0.f, acc1 = 0.f, acc2 = 0.f, dS = 0.f, dS2 = 0.f;

  for (int t = waveId; t < TILES; t += numWaves) {
    int mode = t / TILES1;
    int rem = t - mode * TILES1;
    int ti = rem / NTILE, tj = rem - ti * NTILE;
    const float *A4, *B4, *wA, *wB;
    float coef;
    if (mode == 0)      { A4 = W+OFF_A4C_B; B4 = W+OFF_B4C_B; wA = W+OFF_W_B; wB = W+OFF_W_B; coef =  1.f; }
    else if (mode == 1) { A4 = W+OFF_A4C_T; B4 = W+OFF_B4C_T; wA = W+OFF_W_T; wB = W+OFF_W_T; coef =  1.f; }
    else                { A4 = W+OFF_A4C_B; B4 = W+OFF_B4C_T; wA = W+OFF_W_B; wB = W+OFF_W_T; coef = -2.f; }
    int i0 = ti << 4, j0 = tj << 4;

    // A frag (16x4 f32, 2 VGPRs): lanes<16 hold K=0,1 of row ln; lanes>=16 hold K=2,3
    const float* ar = A4 + ((i0 + ln) << 2) + (hf << 1);
    v2f a; a.x = ar[0]; a.y = ar[1];
    // B frag (4x16 f32, 2 VGPRs): v0 = rows K=0 (lanes<16) / K=1 (lanes>=16); v1 = K=2 / K=3
    const float* br = B4 + ((j0 + ln) << 2) + hf;
    v2f b; b.x = br[0]; b.y = br[2];
    v8f c = {};
    c = __builtin_amdgcn_wmma_f32_16x16x4_f32(false, a, false, b, (short)0, c, false, false);

    float wj = wB[j0 + ln] * coef;
    float t0 = 0.f, t1 = 0.f, t2 = 0.f;
    #pragma unroll
    for (int r = 0; r < 8; ++r) {
      // C/D layout: lanes<16 -> M=r ; lanes>=16 -> M=8+r ; N = lane&15
      float e = fmaxf(c[r], 0.f);
      float wm = wA[i0 + (hf << 3) + r];
      t0 += wm * __builtin_amdgcn_exp2f(-e * c0);
      t1 += wm * __builtin_amdgcn_exp2f(-e * c1);
      t2 += wm * __builtin_amdgcn_exp2f(-e * c2);
    }
    acc0 += wj * t0; acc1 += wj * t1; acc2 += wj * t2;

    if (mode == 2) {  // uncentered distances for gating stats
      const float* aru = W + OFF_A4U_B + ((i0 + ln) << 2) + (hf << 1);
      v2f au; au.x = aru[0]; au.y = aru[1];
      const float* bru = W + OFF_B4U_T + ((j0 + ln) << 2) + hf;
      v2f bu; bu.x = bru[0]; bu.y = bru[2];
      v8f cu = {};
      cu = __builtin_amdgcn_wmma_f32_16x16x4_f32(false, au, false, bu, (short)0, cu, false, false);
      #pragma unroll
      for (int r = 0; r < 8; ++r) {
        float e = fmaxf(cu[r], 0.f);
        dS  += __builtin_amdgcn_sqrtf(e);
        dS2 += e;
      }
    }
  }

  for (int o = 16; o > 0; o >>= 1) {
    acc0 += __shfl_down(acc0, o); acc1 += __shfl_down(acc1, o);
    acc2 += __shfl_down(acc2, o);
    dS += __shfl_down(dS, o); dS2 += __shfl_down(dS2, o);
  }
  if (lane == 0) {
    atomicAdd(&W[OFF_BAND + 0], acc0);
    atomicAdd(&W[OFF_BAND + 1], acc1);
    atomicAdd(&W[OFF_BAND + 2], acc2);
    atomicAdd((double*)W + 0, (double)dS);
    atomicAdd((double*)W + 1, (double)dS2);
  }
}

__global__ void k_final(float* W,
                        const float* __restrict__ gw1, const float* __restrict__ gb1,
                        const float* __restrict__ gw2, const float* __restrict__ gb2,
                        const float* __restrict__ bias, float* out) {
  if (threadIdx.x != 0) return;
  double* D = (double*)W;
  double NM = (double)NPTS * (double)NPTS;
  double mean = D[0] / NM;
  double var = (D[1] - NM * mean * mean) / (NM - 1.0);
  float st[4] = {(float)mean, (float)var, 0.f, W[OFF_PARAMS + 6]};
  float lg0 = gb2[0], lg1 = gb2[1], lg2 = gb2[2];
  for (int k = 0; k < 32; ++k) {
    float h = gb1[k];
    for (int d = 0; d < 4; ++d) h += st[d] * gw1[d * 32 + k];
    h = fmaxf(h, 0.f);
    lg0 += h * gw2[k * 3 + 0];
    lg1 += h * gw2[k * 3 + 1];
    lg2 += h * gw2[k * 3 + 2];
  }
  float g0 = fsoftplus(lg0), g1 = fsoftplus(lg1), g2 = fsoftplus(lg2);
  float gs = g0 + g1 + g2;
  out[0] = (g0 * W[OFF_BAND] + g1 * W[OFF_BAND+1] + g2 * W[OFF_BAND+2]) / gs + bias[0];
}

extern "C" void kernel_launch(void* const* d_in, const int* in_sizes, int n_in,
                              void* d_out, int out_size, void* d_ws, size_t ws_size,
                              hipStream_t stream) {
  const float* base       = (const float*)d_in[0];
  const float* targ       = (const float*)d_in[1];
  const float* log_sigmas = (const float*)d_in[2];
  const float* log_scale  = (const float*)d_in[3];
  const float* wn_w1      = (const float*)d_in[4];
  const float* wn_b1      = (const float*)d_in[5];
  const float* wn_w2      = (const float*)d_in[6];
  const float* wn_b2      = (const float*)d_in[7];
  const float* g_w1       = (const float*)d_in[8];
  const float* g_b1       = (const float*)d_in[9];
  const float* g_w2       = (const float*)d_in[10];
  const float* g_b2       = (const float*)d_in[11];
  const float* bias       = (const float*)d_in[12];
  float* W = (float*)d_ws;

  k_init<<<1, 32, 0, stream>>>(W);
  k_prep<<<(2 * NPTS) / 256, 256, 0, stream>>>(base, targ, log_scale,
                                               wn_w1, wn_b1, wn_w2, wn_b2, W);
  k_means<<<1, 1, 0, stream>>>(W);
  k_build<<<(2 * NPTS) / 256, 256, 0, stream>>>(W);
  k_tiles<<<3072, 256, 0, stream>>>(W, log_sigmas);
  k_final<<<1, 1, 0, stream>>>(W, g_w1, g_b1, g_w2, g_b2, bias, (float*)d_out);
}